// RGCNLayer_55233279426724
// MI455X (gfx1250) — compile-verified
//
#include <hip/hip_runtime.h>

typedef __attribute__((ext_vector_type(2))) float v2f;
typedef __attribute__((ext_vector_type(8))) float v8f;

#define RELS 16
#define BASES 8
#define FDIM 64            // IN == OUT == 64
#define STRIP_TILES 8      // tiles of 16 edges per block strip
#define STRIP_EDGES 128    // 8 * 16

// ---------------------------------------------------------------------------
// Kernel 1: build W[r,i,o] replicating the reference's reinterpreting views:
//   V = weight.view(IN,B,OUT); M = einsum('rb,ibo->iro', w_comp, V); W = M.view(R,IN,OUT)
// Index algebra:
//   W[r,i,o] = sum_b w_comp[i&15, b] * weight[ im>>3, (im&7)*8 + b, o ],  im = 4r + (i>>4)
// ---------------------------------------------------------------------------
__global__ void rgcn_build_w(const float* __restrict__ weight,
                             const float* __restrict__ w_comp,
                             float* __restrict__ Wf) {
    int tid = blockIdx.x * blockDim.x + threadIdx.x;     // 16*64*64 = 65536
    if (tid >= RELS * FDIM * FDIM) return;
    int r = tid >> 12;
    int i = (tid >> 6) & 63;
    int o = tid & 63;
    int im = 4 * r + (i >> 4);
    int rm = i & 15;
    const float* wbase = weight + (size_t)(im >> 3) * FDIM * FDIM + (size_t)((im & 7) * 8) * FDIM + o;
    const float* cbase = w_comp + rm * BASES;
    float acc = 0.f;
#pragma unroll
    for (int b = 0; b < BASES; ++b)
        acc += cbase[b] * wbase[(size_t)b * FDIM];
    Wf[tid] = acc;
}

// ---------------------------------------------------------------------------
// Kernels 2-4: bin edges by relation, materializing binned src/dst node ids
// ---------------------------------------------------------------------------
__global__ void rgcn_zero_counts(int* __restrict__ counts) {
    if (threadIdx.x < RELS) counts[threadIdx.x] = 0;
}

__global__ void rgcn_histo(const int* __restrict__ rel, int E, int* __restrict__ counts) {
    __shared__ int lc[RELS];
    if (threadIdx.x < RELS) lc[threadIdx.x] = 0;
    __syncthreads();
    int i = blockIdx.x * blockDim.x + threadIdx.x;
    if (i < E) atomicAdd(&lc[rel[i]], 1);
    __syncthreads();
    if (threadIdx.x < RELS && lc[threadIdx.x]) atomicAdd(&counts[threadIdx.x], lc[threadIdx.x]);
}

__global__ void rgcn_prefix(const int* __restrict__ counts, int* __restrict__ offsets,
                            int* __restrict__ cursors, int* __restrict__ stripOff) {
    if (threadIdx.x == 0) {
        int acc = 0, sacc = 0;
        for (int r = 0; r < RELS; ++r) {
            offsets[r] = acc;
            cursors[r] = acc;
            stripOff[r] = sacc;
            acc  += counts[r];
            sacc += (counts[r] + STRIP_EDGES - 1) / STRIP_EDGES;
        }
        offsets[RELS]  = acc;
        stripOff[RELS] = sacc;
    }
}

// Directly scatter src/dst node ids into relation bins (no edge-id indirection
// left for the hot kernel).
__global__ void rgcn_scatter_ids(const int* __restrict__ rel, const int* __restrict__ src,
                                 const int* __restrict__ dst, int E,
                                 int* __restrict__ cursors,
                                 int* __restrict__ srcS, int* __restrict__ dstS) {
    __shared__ int lc[RELS], lbase[RELS];
    if (threadIdx.x < RELS) lc[threadIdx.x] = 0;
    __syncthreads();
    int i = blockIdx.x * blockDim.x + threadIdx.x;
    int r = 0, mypos = 0;
    bool act = (i < E);
    if (act) { r = rel[i]; mypos = atomicAdd(&lc[r], 1); }
    __syncthreads();
    if (threadIdx.x < RELS)
        lbase[threadIdx.x] = lc[threadIdx.x] ? atomicAdd(&cursors[threadIdx.x], lc[threadIdx.x]) : 0;
    __syncthreads();
    if (act) {
        int slot = lbase[r] + mypos;
        srcS[slot] = src[i];
        dstS[slot] = dst[i];
    }
}

// ---------------------------------------------------------------------------
// Kernel 5: h_new = relu(h); h2 = 0
// ---------------------------------------------------------------------------
__global__ void rgcn_relu_zero(const float* __restrict__ h, float* __restrict__ out,
                               float* __restrict__ h2, int n) {
    int i = blockIdx.x * blockDim.x + threadIdx.x;
    if (i < n) {
        float x = h[i];
        out[i] = x > 0.f ? x : 0.f;
        h2[i]  = 0.f;
    }
}

// ---------------------------------------------------------------------------
// Kernel 6: strip-processing message GEMM + scatter-add. No LDS, no barriers.
// One block = one strip of up to 128 edges (8 tiles of 16) of one relation.
// 4 waves; wave w owns output columns [16w, 16w+16) for the whole strip.
// B operand (16 cols x 64 K of W[r]) lives in 32 VGPRs for the entire strip.
// Per tile: dst node ids prefetched into registers BEFORE the WMMA chain (their
// latency hides under compute); A = 16 gathered h rows (single indirection);
// 16x V_WMMA_F32_16X16X4_F32; f32 atomic scatter-add into h2[dst] (L2-resident).
// ---------------------------------------------------------------------------
__global__ __launch_bounds__(128) void rgcn_msg_wmma(
    const float* __restrict__ h, const float* __restrict__ Wf,
    const int* __restrict__ srcS, const int* __restrict__ dstS,
    const int* __restrict__ counts, const int* __restrict__ offsets,
    const int* __restrict__ stripOff, float* __restrict__ h2) {

    int strip = blockIdx.x;
    if (strip >= stripOff[RELS]) return;       // block-uniform exit
    int r = 0;
    while (strip >= stripOff[r + 1]) ++r;      // <=16 iterations, wave-uniform
    int s = strip - stripOff[r];

    int lane = threadIdx.x & 31;
    int wave = threadIdx.x >> 5;
    int o0   = wave * 16;
    int n    = lane & 15;
    int half = lane >> 4;
    int kb   = half * 2;                       // K sub-offset of this lane half

    // B fragments for this wave's 16 output columns, all 64 K, in registers:
    //   bf[kk][v] = W[r, 4*kk + kb + v, o0 + n]
    const float* Wr = Wf + (size_t)r * FDIM * FDIM + o0 + n;
    v2f bf[16];
#pragma unroll
    for (int kk = 0; kk < 16; ++kk) {
        bf[kk].x = Wr[(size_t)(4 * kk + kb) * FDIM];
        bf[kk].y = Wr[(size_t)(4 * kk + kb + 1) * FDIM];
    }

    int cnt   = counts[r];
    int base0 = offsets[r] + s * STRIP_EDGES;
    int rem0  = cnt - s * STRIP_EDGES;         // > 0 by construction

    for (int t = 0; t < STRIP_TILES; ++t) {
        int rem = rem0 - t * 16;
        if (rem <= 0) break;                   // wave-uniform: EXEC stays all-1
        int base = base0 + t * 16;

        // A row for this lane: edge (l&15) of the tile, gathered from h.
        bool aval = (n < rem);
        int  srow = aval ? srcS[base + n] : 0;
        const float* hrow = h + (size_t)srow * FDIM;

        // Prefetch scatter targets (rows m = v + 8*half) before the WMMA chain.
        int dnode[8];
#pragma unroll
        for (int v = 0; v < 8; ++v) {
            int m = v + half * 8;
            dnode[v] = (m < rem) ? dstS[base + m] : -1;   // same addr across half-wave
        }

        v8f c = {};
#pragma unroll
        for (int kk = 0; kk < 16; ++kk) {
            float2 av = *reinterpret_cast<const float2*>(hrow + 4 * kk + kb);
            v2f a; a.x = av.x; a.y = av.y;
            c = __builtin_amdgcn_wmma_f32_16x16x4_f32(
                    /*neg_a=*/false, a, /*neg_b=*/false, bf[kk],
                    /*c_mod=*/(short)0, c, /*reuse_a=*/false, /*reuse_b=*/false);
        }

        // D layout: row m = v + 8*half, col = o0 + n. Scatter-add to h2[dst].
#pragma unroll
        for (int v = 0; v < 8; ++v) {
            if (dnode[v] >= 0) {
                atomicAdd(&h2[(size_t)dnode[v] * FDIM + o0 + n], c[v]);
            }
        }
    }
}

// ---------------------------------------------------------------------------
extern "C" void kernel_launch(void* const* d_in, const int* in_sizes, int n_in,
                              void* d_out, int out_size, void* d_ws, size_t ws_size,
                              hipStream_t stream) {
    const float* h      = (const float*)d_in[0];
    const float* weight = (const float*)d_in[1];
    const float* w_comp = (const float*)d_in[2];
    const int*   src    = (const int*)d_in[3];
    const int*   dst    = (const int*)d_in[4];
    const int*   rel    = (const int*)d_in[5];

    const int N = in_sizes[0] / FDIM;
    const int E = in_sizes[3];

    float* out_h  = (float*)d_out;                 // relu(h): N*64
    float* out_h2 = out_h + (size_t)N * FDIM;      // h2:      N*64

    // Workspace layout (~8.3 MB total for E=1M)
    float* Wf    = (float*)d_ws;                                   // 16*64*64 floats
    int*   meta  = (int*)((char*)d_ws + (size_t)RELS * FDIM * FDIM * sizeof(float));
    int*   counts   = meta;              // 16
    int*   offsets  = meta + 16;         // 17
    int*   cursors  = meta + 33;         // 16
    int*   stripOff = meta + 64;         // 17
    int*   srcS     = meta + 96;         // E ints (relation-binned src node ids)
    int*   dstS     = srcS + E;          // E ints (relation-binned dst node ids)

    // 1) basis combination -> W
    rgcn_build_w<<<(RELS * FDIM * FDIM + 255) / 256, 256, 0, stream>>>(weight, w_comp, Wf);
    // 2) bin edges by relation
    rgcn_zero_counts<<<1, 64, 0, stream>>>(counts);
    rgcn_histo<<<(E + 255) / 256, 256, 0, stream>>>(rel, E, counts);
    rgcn_prefix<<<1, 32, 0, stream>>>(counts, offsets, cursors, stripOff);
    rgcn_scatter_ids<<<(E + 255) / 256, 256, 0, stream>>>(rel, src, dst, E, cursors, srcS, dstS);
    // 3) relu(h) and zero h2
    rgcn_relu_zero<<<((N * FDIM) + 255) / 256, 256, 0, stream>>>(h, out_h, out_h2, N * FDIM);
    // 4) WMMA message GEMM + scatter-add (upper bound on strips; extras exit early)
    int maxStrips = E / STRIP_EDGES + RELS;
    rgcn_msg_wmma<<<maxStrips, 128, 0, stream>>>(h, Wf, srcS, dstS,
                                                 counts, offsets, stripOff, out_h2);
}